// SparseLinear_50560355008948
// MI455X (gfx1250) — compile-verified
//
#include <hip/hip_runtime.h>

// Problem constants (from reference)
#define BATCH      16384
#define OUTF       512
#define INF        40960
#define NNZ_TOT    524288
#define TILE_ROWS  16
#define ROW_SENTINEL 0x40000000

typedef float v2f __attribute__((ext_vector_type(2)));
typedef float v8f __attribute__((ext_vector_type(8)));

// ---------------- Pass 0: zero counts + cursors, write sentinel pad ----------------
__global__ void sl_zero_kernel(int* __restrict__ counts, int* __restrict__ cursor,
                               int* __restrict__ s_feat, int* __restrict__ s_row,
                               float* __restrict__ s_val) {
    int i = blockIdx.x * blockDim.x + threadIdx.x;
    if (i < BATCH) { counts[i] = 0; cursor[i] = 0; }
    if (i < 4) {                       // 4-entry tail pad: never matches any tile row
        s_row[NNZ_TOT + i]  = ROW_SENTINEL;
        s_feat[NNZ_TOT + i] = 0;
        s_val[NNZ_TOT + i]  = 0.0f;
    }
}

// ---------------- Pass 1: histogram of batch rows ----------------
__global__ void sl_hist_kernel(const long long* __restrict__ batch_idx,
                               int* __restrict__ counts) {
    int i = blockIdx.x * blockDim.x + threadIdx.x;
    if (i < NNZ_TOT) atomicAdd(&counts[(int)batch_idx[i]], 1);
}

// ---------------- Pass 2: exclusive scan of 16384 counts (1 workgroup) ----------------
__global__ __launch_bounds__(1024)
void sl_scan_kernel(const int* __restrict__ counts, int* __restrict__ row_start) {
    __shared__ int sm[1024];
    const int t = threadIdx.x;
    const int base = t * 16;
    int local[16];
    int sum = 0;
#pragma unroll
    for (int i = 0; i < 16; ++i) { local[i] = sum; sum += counts[base + i]; }
    sm[t] = sum;
    __syncthreads();
    for (int off = 1; off < 1024; off <<= 1) {
        int v = (t >= off) ? sm[t - off] : 0;
        __syncthreads();
        sm[t] += v;
        __syncthreads();
    }
    const int excl = (t == 0) ? 0 : sm[t - 1];
#pragma unroll
    for (int i = 0; i < 16; ++i) row_start[base + i] = excl + local[i];
    if (t == 1023) row_start[BATCH] = sm[1023];
}

// ---------------- Pass 3: scatter nnz into row-sorted order ----------------
__global__ void sl_scatter_kernel(const long long* __restrict__ batch_idx,
                                  const long long* __restrict__ feat_idx,
                                  const float* __restrict__ values,
                                  const int* __restrict__ row_start,
                                  int* __restrict__ cursor,
                                  int* __restrict__ s_feat,
                                  int* __restrict__ s_row,
                                  float* __restrict__ s_val) {
    int i = blockIdx.x * blockDim.x + threadIdx.x;
    if (i >= NNZ_TOT) return;
    int b = (int)batch_idx[i];
    int pos = row_start[b] + atomicAdd(&cursor[b], 1);
    s_feat[pos] = (int)feat_idx[i];
    s_row[pos]  = b;
    s_val[pos]  = values[i];
}

// ---------------- Pass 4: per-tile scatter-GEMM via V_WMMA_F32_16X16X4_F32 ----------------
// Block = 512 threads (16 waves) owns output rows [tile*16, tile*16+16) x 512 cols.
// Wave w owns cols [w*32, w*32+32) as two 16x16 f32 accumulators.
// K-step consumes 4 sorted nnz: A[16x4] = one-hot(row)*value, B[4x16] = gathered
// weight slices, D = A*B + C.  Branch-free: loop start aligned down to 4 (leading
// entries are the previous tile's rows -> A==0) and the sorted arrays carry a
// 4-entry sentinel tail (row=ROW_SENTINEL -> A==0), so no bounds checks and all
// metadata reads are even-indexed 64-bit loads.
__global__ __launch_bounds__(512)
void sl_spmm_wmma_kernel(const int* __restrict__ row_start,
                         const int* __restrict__ s_feat,
                         const int* __restrict__ s_row,
                         const float* __restrict__ s_val,
                         const float* __restrict__ weight,
                         const float* __restrict__ bias,
                         float* __restrict__ out) {
    const int tile = blockIdx.x;
    const int row0 = tile * TILE_ROWS;
    const int jBeg = row_start[row0] & ~3;          // align down: extra rows can't match
    const int jEnd = row_start[row0 + TILE_ROWS];   // uniform across block

    const int lane = threadIdx.x & 31;
    const int half = lane >> 4;                     // 0 or 1
    const int m    = lane & 15;                     // M for A, N for B/C/D
    const int k0   = half * 2;                      // this lane holds K = k0, k0+1
    const int col0 = (threadIdx.x >> 5) * 32;

    const int2*   s_row2  = (const int2*)s_row;
    const int2*   s_feat2 = (const int2*)s_feat;
    const float2* s_val2  = (const float2*)s_val;

    v8f c0 = {};
    v8f c1 = {};

    for (int j = jBeg; j < jEnd; j += 4) {
        const int p = (j + k0) >> 1;                // even index -> b64 loads
        const int2   r2 = s_row2[p];
        const int2   f2 = s_feat2[p];
        const float2 v2 = s_val2[p];

        v2f a;
        a.x = (r2.x - row0 == m) ? v2.x : 0.0f;     // single v_cndmask each
        a.y = (r2.y - row0 == m) ? v2.y : 0.0f;

        const int w0 = f2.x * OUTF + col0 + m;      // 32-bit element offsets
        const int w1 = f2.y * OUTF + col0 + m;
        v2f b0, b1;
        b0.x = weight[w0];        b0.y = weight[w1];
        b1.x = weight[w0 + 16];   b1.y = weight[w1 + 16];

        c0 = __builtin_amdgcn_wmma_f32_16x16x4_f32(false, a, false, b0, (short)0, c0, false, false);
        c1 = __builtin_amdgcn_wmma_f32_16x16x4_f32(false, a, false, b1, (short)0, c1, false, false);
    }

    // C/D layout: VGPR v, lanes 0-15 -> M=v, lanes 16-31 -> M=v+8; N = lane&15.
    const float bb0 = bias[col0 + m];
    const float bb1 = bias[col0 + 16 + m];
#pragma unroll
    for (int v = 0; v < 8; ++v) {
        const int r = row0 + v + half * 8;
        out[(size_t)r * OUTF + col0 + m]      = c0[v] + bb0;
        out[(size_t)r * OUTF + col0 + 16 + m] = c1[v] + bb1;
    }
}

extern "C" void kernel_launch(void* const* d_in, const int* in_sizes, int n_in,
                              void* d_out, int out_size, void* d_ws, size_t ws_size,
                              hipStream_t stream) {
    // Inputs (reference order): indices[2,NNZ] int64, values[NNZ] f32, batch_size,
    //                           weight[IN*OUT] f32, bias[OUT] f32
    const long long* indices = (const long long*)d_in[0];
    const long long* batch_i = indices;             // indices[0, :]
    const long long* feat_i  = indices + NNZ_TOT;   // indices[1, :]
    const float*     values  = (const float*)d_in[1];
    const float*     weight  = (const float*)d_in[3];
    const float*     bias    = (const float*)d_in[4];
    float*           out     = (float*)d_out;

    // Workspace layout (~6.5 MB); each sorted array has a 4-entry sentinel tail.
    int*   counts = (int*)d_ws;                       // BATCH
    int*   cursor = counts + BATCH;                   // BATCH
    int*   rstart = cursor + BATCH;                   // BATCH+1 (padded to +8)
    int*   s_feat = rstart + (BATCH + 8);             // NNZ+4  (8B-aligned)
    int*   s_row  = s_feat + (NNZ_TOT + 4);           // NNZ+4
    float* s_val  = (float*)(s_row + (NNZ_TOT + 4));  // NNZ+4

    sl_zero_kernel<<<BATCH / 256, 256, 0, stream>>>(counts, cursor, s_feat, s_row, s_val);
    sl_hist_kernel<<<NNZ_TOT / 256, 256, 0, stream>>>(batch_i, counts);
    sl_scan_kernel<<<1, 1024, 0, stream>>>(counts, rstart);
    sl_scatter_kernel<<<NNZ_TOT / 256, 256, 0, stream>>>(batch_i, feat_i, values,
                                                         rstart, cursor,
                                                         s_feat, s_row, s_val);
    sl_spmm_wmma_kernel<<<BATCH / TILE_ROWS, 512, 0, stream>>>(rstart, s_feat, s_row,
                                                               s_val, weight, bias, out);
}